// MixHop_65274912964848
// MI455X (gfx1250) — compile-verified
//
#include <hip/hip_runtime.h>

typedef float v2f __attribute__((ext_vector_type(2)));
typedef float v8f __attribute__((ext_vector_type(8)));

#define N_NODES 50000
#define N_EDGES 800000
#define IN_DIM  128
#define OUT_DIM 64

// ---------------------------------------------------------------------------
// 1. zero workspace region (px accumulator + degree counters)
// ---------------------------------------------------------------------------
__global__ void mixhop_zero_kernel(float* __restrict__ p, int n) {
    int i = blockIdx.x * blockDim.x + threadIdx.x;
    if (i < n) p[i] = 0.0f;
}

// ---------------------------------------------------------------------------
// 2. in-degree of col  (edge_index flat layout: [0..E) = row, [E..2E) = col)
// ---------------------------------------------------------------------------
__global__ void mixhop_degree_kernel(const long long* __restrict__ ei,
                                     unsigned int* __restrict__ deg) {
    int e = blockIdx.x * blockDim.x + threadIdx.x;
    if (e < N_EDGES) {
        int c = (int)ei[N_EDGES + e];
        atomicAdd(&deg[c], 1u);
    }
}

// ---------------------------------------------------------------------------
// 3. dis[n] = deg>0 ? 1/sqrt(deg) : 0
// ---------------------------------------------------------------------------
__global__ void mixhop_dis_kernel(const unsigned int* __restrict__ deg,
                                  float* __restrict__ dis) {
    int n = blockIdx.x * blockDim.x + threadIdx.x;
    if (n < N_NODES) {
        unsigned int d = deg[n];
        dis[n] = d ? rsqrtf((float)d) : 0.0f;
    }
}

// ---------------------------------------------------------------------------
// 4. fold weights:  A0 = F0*W0,  A1 = F1*W1 + F2*W2   (each [64,128])
//    Wfc[o, j] row-major [64,384]; Wk[j, i] row-major [128,128]
// ---------------------------------------------------------------------------
__global__ void mixhop_fold_kernel(const float* __restrict__ W0,
                                   const float* __restrict__ W1,
                                   const float* __restrict__ W2,
                                   const float* __restrict__ Wfc,
                                   float* __restrict__ A0,
                                   float* __restrict__ A1) {
    int t = blockIdx.x * blockDim.x + threadIdx.x;
    if (t >= OUT_DIM * IN_DIM) return;
    int o = t >> 7;          // out channel
    int i = t & 127;         // in  channel
    float s0 = 0.0f, s1 = 0.0f;
    const float* fr = Wfc + (size_t)o * (3 * IN_DIM);
    for (int j = 0; j < 128; ++j) {
        s0 += fr[j]       * W0[j * 128 + i];
        s1 += fr[128 + j] * W1[j * 128 + i];
        s1 += fr[256 + j] * W2[j * 128 + i];
    }
    A0[t] = s0;
    A1[t] = s1;
}

// ---------------------------------------------------------------------------
// 5. edge scatter: px[col] += norm * x[row].  One wave32 per edge,
//    each lane handles one float4 (32 lanes x 4 = 128 features).
// ---------------------------------------------------------------------------
__global__ void mixhop_scatter_kernel(const float* __restrict__ x,
                                      const long long* __restrict__ ei,
                                      const float* __restrict__ dis,
                                      float* __restrict__ px) {
    int gid  = blockIdx.x * blockDim.x + threadIdx.x;
    int e    = gid >> 5;
    int lane = gid & 31;
    if (e >= N_EDGES) return;
    int r = (int)ei[e];
    int c = (int)ei[N_EDGES + e];
    float norm = dis[r] * dis[c];
    float4 v = ((const float4*)(x + (size_t)r * IN_DIM))[lane];
    float* dst = px + (size_t)c * IN_DIM + lane * 4;
    unsafeAtomicAdd(dst + 0, v.x * norm);
    unsafeAtomicAdd(dst + 1, v.y * norm);
    unsafeAtomicAdd(dst + 2, v.z * norm);
    unsafeAtomicAdd(dst + 3, v.w * norm);
}

// ---------------------------------------------------------------------------
// 6. WMMA GEMM:  out[N,64] = x[N,128]*A0^T + px[N,128]*A1^T
//    8 waves/block; wave (rt, ct) -> 16-node x 16-outcol tile.
//    V_WMMA_F32_16X16X4_F32 fragment layout (ISA 7.12.2):
//      A 16x4:  lanes 0-15 hold M=lane, K=0,1 ; lanes 16-31 hold M=lane-16, K=2,3
//      B 4x16:  lanes 0-15 hold N=lane, K=0,1 ; lanes 16-31 hold N=lane-16, K=2,3
//      C/D:     VGPR r -> (M = r + 8*hi, N = lane&15)
// ---------------------------------------------------------------------------
__global__ __launch_bounds__(256)
void mixhop_gemm_kernel(const float* __restrict__ x,
                        const float* __restrict__ px,
                        const float* __restrict__ A0,
                        const float* __restrict__ A1,
                        float* __restrict__ out) {
    __shared__ float sA0[OUT_DIM * IN_DIM];   // 32 KB
    __shared__ float sA1[OUT_DIM * IN_DIM];   // 32 KB
    const int tid = threadIdx.x;
    for (int i = tid; i < OUT_DIM * IN_DIM; i += 256) {
        sA0[i] = A0[i];
        sA1[i] = A1[i];
    }
    __syncthreads();

    const int wave     = tid >> 5;
    const int lane     = tid & 31;
    const int row_tile = wave >> 2;                 // 0..1
    const int col_tile = wave & 3;                  // 0..3
    const int row_base = blockIdx.x * 32 + row_tile * 16;
    const int col_base = col_tile * 16;

    const int lm = lane & 15;
    const int hi = lane >> 4;                       // 0: K=0,1  1: K=2,3

    int row = row_base + lm;
    if (row >= N_NODES) row = N_NODES - 1;          // clamp loads; stores guarded

    const float* xr  = x   + (size_t)row * IN_DIM + hi * 2;
    const float* pxr = px  + (size_t)row * IN_DIM + hi * 2;
    const float* b0  = sA0 + (size_t)(col_base + lm) * IN_DIM + hi * 2;
    const float* b1  = sA1 + (size_t)(col_base + lm) * IN_DIM + hi * 2;

    v8f acc = {};
#pragma unroll
    for (int k = 0; k < IN_DIM; k += 4) {
        v2f a = *(const v2f*)(xr + k);
        v2f b = *(const v2f*)(b0 + k);
        acc = __builtin_amdgcn_wmma_f32_16x16x4_f32(
            false, a, false, b, (short)0, acc, false, false);
    }
#pragma unroll
    for (int k = 0; k < IN_DIM; k += 4) {
        v2f a = *(const v2f*)(pxr + k);
        v2f b = *(const v2f*)(b1 + k);
        acc = __builtin_amdgcn_wmma_f32_16x16x4_f32(
            false, a, false, b, (short)0, acc, false, false);
    }

    const int n  = col_base + lm;
    const int m0 = row_base + hi * 8;
#pragma unroll
    for (int r = 0; r < 8; ++r) {
        int m = m0 + r;
        if (m < N_NODES) out[(size_t)m * OUT_DIM + n] = acc[r];
    }
}

// ---------------------------------------------------------------------------
// launch
// ---------------------------------------------------------------------------
extern "C" void kernel_launch(void* const* d_in, const int* in_sizes, int n_in,
                              void* d_out, int out_size, void* d_ws, size_t ws_size,
                              hipStream_t stream) {
    const float*     x   = (const float*)d_in[0];
    const long long* ei  = (const long long*)d_in[1];   // int64 per reference
    const float*     W0  = (const float*)d_in[2];
    const float*     W1  = (const float*)d_in[3];
    const float*     W2  = (const float*)d_in[4];
    const float*     Wfc = (const float*)d_in[5];
    float*           out = (float*)d_out;

    // workspace layout (floats): px | deg(u32) | dis | A0 | A1   ~26 MB
    float*        ws  = (float*)d_ws;
    float*        px  = ws;                                   // 6,400,000
    unsigned int* deg = (unsigned int*)(ws + 6400000);        //    50,000
    float*        dis = ws + 6400000 + 50048;                 //    50,000
    float*        A0  = dis + 50048;                          //     8,192
    float*        A1  = A0 + OUT_DIM * IN_DIM;                //     8,192

    // 1. zero px + deg (contiguous region)
    {
        int n = 6400000 + 50048;
        mixhop_zero_kernel<<<(n + 255) / 256, 256, 0, stream>>>(px, n);
    }
    // 2. degrees
    mixhop_degree_kernel<<<(N_EDGES + 255) / 256, 256, 0, stream>>>(ei, deg);
    // 3. inverse sqrt degrees
    mixhop_dis_kernel<<<(N_NODES + 255) / 256, 256, 0, stream>>>(deg, dis);
    // 4. fold dense weights
    mixhop_fold_kernel<<<(OUT_DIM * IN_DIM + 255) / 256, 256, 0, stream>>>(
        W0, W1, W2, Wfc, A0, A1);
    // 5. edge scatter (1 wave per edge)
    {
        long long threads = (long long)N_EDGES * 32;
        mixhop_scatter_kernel<<<(unsigned)((threads + 255) / 256), 256, 0, stream>>>(
            x, ei, dis, px);
    }
    // 6. fused WMMA GEMM (32 nodes per block)
    mixhop_gemm_kernel<<<(N_NODES + 31) / 32, 256, 0, stream>>>(x, px, A0, A1, out);
}